// Decoder_5325759447139
// MI455X (gfx1250) — compile-verified
//
#include <hip/hip_runtime.h>
#include <hip/hip_bf16.h>
#include <math.h>

// ---------------- problem constants (from reference) ----------------
constexpr int kV    = 32000;
constexpr int kE    = 512;
constexpr int kH    = 1024;
constexpr int kENC  = 1024;
constexpr int kSTY  = 128;
constexpr int kB    = 32;
constexpr int kSENC = 64;
constexpr int kT    = 64;
constexpr int kDIN  = kE + kSTY + kENC;   // 1664
constexpr int kH3   = 3 * kH;             // 3072
constexpr int kKH0  = kENC + kSTY;        // 1152

typedef __attribute__((ext_vector_type(2))) float v2f;
typedef __attribute__((ext_vector_type(8))) float v8f;

// ---------------------------------------------------------------
// fp32 WMMA GEMM, 2x2 register-blocked:
//   C[M x N] = A[M x K](row stride lda) @ W[N x K]^T + bias[N]
// One wave computes a 32x32 block of C (four 16x16 WMMA tiles).
// Per K-step of 4: 2 A-fragment loads + 2 B-fragment loads feed 4
// V_WMMA_F32_16X16X4_F32 -> 1 load per WMMA (8 FLOP/byte from L2).
// Fragment layout (wave32, ISA 7.12.2):
//   A (16x4):  lanes 0-15 -> M=lane, VGPR0=K0,VGPR1=K1 ; lanes 16-31 -> K2,K3
//   B (4x16):  lanes 0-15 -> N=lane, rows K0,K1        ; lanes 16-31 -> rows K2,K3
//   C/D:       VGPR j -> row (j + 8*half), col = lane&15
// Requires M % 32 == 0, N % 32 == 0, K % 4 == 0 (true for all uses here).
// Early-exit is wave-uniform so EXEC stays all-ones for WMMA.
// ---------------------------------------------------------------
__global__ void gemm_wmma_f32(const float* __restrict__ A,
                              const float* __restrict__ W,
                              const float* __restrict__ bias,
                              float* __restrict__ C,
                              int M, int K, int N, int lda) {
    const int wave = (blockIdx.x * blockDim.x + threadIdx.x) >> 5;
    const int lane = threadIdx.x & 31;
    const int blkN = N >> 5;                   // 32-wide column blocks
    const int blks = (M >> 5) * blkN;
    if (wave >= blks) return;                  // uniform per wave

    const int bm   = wave / blkN;
    const int bn   = wave - bm * blkN;
    const int half = lane >> 4;                // 0: K{0,1}, 1: K{2,3}
    const int l16  = lane & 15;

    const int rowA0 = bm * 32 + l16;           // M rows this lane feeds
    const int rowA1 = rowA0 + 16;
    const int colN0 = bn * 32 + l16;           // N cols this lane feeds/writes
    const int colN1 = colN0 + 16;

    const float* __restrict__ Arow0 = A + (size_t)rowA0 * lda;
    const float* __restrict__ Arow1 = A + (size_t)rowA1 * lda;
    const float* __restrict__ Wrow0 = W + (size_t)colN0 * K;   // W[n][k]
    const float* __restrict__ Wrow1 = W + (size_t)colN1 * K;

    v8f acc00 = {}, acc01 = {}, acc10 = {}, acc11 = {};
    for (int k = 0; k < K; k += 4) {
        const int kk = k + half * 2;
        const float2 a0v = *reinterpret_cast<const float2*>(Arow0 + kk);
        const float2 a1v = *reinterpret_cast<const float2*>(Arow1 + kk);
        const float2 b0v = *reinterpret_cast<const float2*>(Wrow0 + kk);
        const float2 b1v = *reinterpret_cast<const float2*>(Wrow1 + kk);
        v2f a0; a0.x = a0v.x; a0.y = a0v.y;
        v2f a1; a1.x = a1v.x; a1.y = a1v.y;
        v2f b0; b0.x = b0v.x; b0.y = b0v.y;
        v2f b1; b1.x = b1v.x; b1.y = b1v.y;
        acc00 = __builtin_amdgcn_wmma_f32_16x16x4_f32(false, a0, false, b0,
                                                      (short)0, acc00, false, false);
        acc01 = __builtin_amdgcn_wmma_f32_16x16x4_f32(false, a0, false, b1,
                                                      (short)0, acc01, false, false);
        acc10 = __builtin_amdgcn_wmma_f32_16x16x4_f32(false, a1, false, b0,
                                                      (short)0, acc10, false, false);
        acc11 = __builtin_amdgcn_wmma_f32_16x16x4_f32(false, a1, false, b1,
                                                      (short)0, acc11, false, false);
    }

    const float bv0 = bias ? bias[colN0] : 0.0f;
    const float bv1 = bias ? bias[colN1] : 0.0f;
#pragma unroll
    for (int j = 0; j < 8; ++j) {
        const int m0 = bm * 32 + j + 8 * half;         // rows of tiles (0x,.)
        const int m1 = m0 + 16;                        // rows of tiles (1x,.)
        C[(size_t)m0 * N + colN0] = acc00[j] + bv0;
        C[(size_t)m0 * N + colN1] = acc01[j] + bv1;
        C[(size_t)m1 * N + colN0] = acc10[j] + bv0;
        C[(size_t)m1 * N + colN1] = acc11[j] + bv1;
    }
}

// ---------------------------------------------------------------
// Prep: build x = [emb | style | ctx]  (B*T x DIN, row = b*T+t)
//       and a0 = [ctx | style]         (B x (ENC+STY))
// ---------------------------------------------------------------
__global__ void prep_kernel(const float* __restrict__ enc_output,
                            const int* __restrict__ enc_len,
                            const int* __restrict__ styles,
                            const int* __restrict__ dec_in,
                            const float* __restrict__ emb_table,
                            const float* __restrict__ style_table,
                            float* __restrict__ x,
                            float* __restrict__ a0) {
    const size_t totalX  = (size_t)kB * kT * kDIN;
    const size_t totalA0 = (size_t)kB * kKH0;
    const size_t i = (size_t)blockIdx.x * blockDim.x + threadIdx.x;
    if (i < totalX) {
        const int row = (int)(i / kDIN);       // b*T + t
        const int d   = (int)(i - (size_t)row * kDIN);
        const int b   = row / kT;
        float v;
        if (d < kE) {
            v = emb_table[(size_t)dec_in[row] * kE + d];
        } else if (d < kE + kSTY) {
            v = style_table[(size_t)styles[b] * kSTY + (d - kE)];
        } else {
            const int pos = enc_len[b] - 1;
            v = enc_output[((size_t)b * kSENC + pos) * kENC + (d - kE - kSTY)];
        }
        x[i] = v;
    } else if (i < totalX + totalA0) {
        const size_t j = i - totalX;
        const int b = (int)(j / kKH0);
        const int d = (int)(j - (size_t)b * kKH0);
        float v;
        if (d < kENC) {
            const int pos = enc_len[b] - 1;
            v = enc_output[((size_t)b * kSENC + pos) * kENC + d];
        } else {
            v = style_table[(size_t)styles[b] * kSTY + (d - kENC)];
        }
        a0[j] = v;
    }
}

// ---------------------------------------------------------------
// GRU gate fusion for one time step t.
//   r = sig(xr+gr); z = sig(xz+gz); n = tanh(xn + r*gn)
//   h = (1-z)*n + z*h ; also store into hs[:, t, :]
// ---------------------------------------------------------------
__global__ void gru_gate_kernel(const float* __restrict__ x3,
                                const float* __restrict__ g,
                                float* __restrict__ h,
                                float* __restrict__ hs,
                                int t) {
    const int i = blockIdx.x * blockDim.x + threadIdx.x;
    if (i >= kB * kH) return;
    const int b = i / kH;
    const int j = i - b * kH;

    const size_t xbase = ((size_t)b * kT + t) * kH3;
    const float xr = x3[xbase + j];
    const float xz = x3[xbase + kH + j];
    const float xn = x3[xbase + 2 * kH + j];

    const size_t gbase = (size_t)b * kH3;
    const float gr = g[gbase + j];
    const float gz = g[gbase + kH + j];
    const float gn = g[gbase + 2 * kH + j];

    const float r = 1.0f / (1.0f + expf(-(xr + gr)));
    const float z = 1.0f / (1.0f + expf(-(xz + gz)));
    const float n = tanhf(xn + r * gn);

    const float hprev = h[(size_t)b * kH + j];
    const float hnew  = (1.0f - z) * n + z * hprev;
    h[(size_t)b * kH + j] = hnew;
    hs[((size_t)b * kT + t) * kH + j] = hnew;
}

// ---------------------------------------------------------------
// Per-row log-softmax stats + argmax + target gather.
// One 256-thread block per row (b*T+t), V=32000 columns.
// ---------------------------------------------------------------
__global__ void logsoftmax_kernel(const float* __restrict__ logits,
                                  const int* __restrict__ dec_out,
                                  float* __restrict__ lp_out,
                                  float* __restrict__ pred_out) {
    const int row = blockIdx.x;
    const float* __restrict__ lrow = logits + (size_t)row * kV;

    __shared__ float smax[256];
    __shared__ int   sidx[256];
    __shared__ float ssum[256];

    float m = -INFINITY; int mi = kV;
    for (int v = threadIdx.x; v < kV; v += 256) {
        const float val = lrow[v];
        if (val > m || (val == m && v < mi)) { m = val; mi = v; }
    }
    smax[threadIdx.x] = m; sidx[threadIdx.x] = mi;
    __syncthreads();
    for (int s = 128; s > 0; s >>= 1) {
        if ((int)threadIdx.x < s) {
            const float om = smax[threadIdx.x + s];
            const int   oi = sidx[threadIdx.x + s];
            if (om > smax[threadIdx.x] ||
                (om == smax[threadIdx.x] && oi < sidx[threadIdx.x])) {
                smax[threadIdx.x] = om; sidx[threadIdx.x] = oi;
            }
        }
        __syncthreads();
    }
    const float rowmax = smax[0];
    const int   rowarg = sidx[0];

    float sum = 0.0f;
    for (int v = threadIdx.x; v < kV; v += 256) sum += expf(lrow[v] - rowmax);
    ssum[threadIdx.x] = sum;
    __syncthreads();
    for (int s = 128; s > 0; s >>= 1) {
        if ((int)threadIdx.x < s) ssum[threadIdx.x] += ssum[threadIdx.x + s];
        __syncthreads();
    }

    if (threadIdx.x == 0) {
        const float lse = rowmax + logf(ssum[0]);
        const int tgt = dec_out[row];
        lp_out[row]   = lrow[tgt] - lse;
        pred_out[row] = (float)rowarg;
    }
}

// ---------------------------------------------------------------
static inline void launch_gemm(const float* A, const float* W, const float* bias,
                               float* C, int M, int K, int N, int lda,
                               hipStream_t stream) {
    const int blocks32 = (M / 32) * (N / 32);    // one wave per 32x32 C block
    const int wavesPerBlock = 8;                 // 256 threads
    const int grid = (blocks32 + wavesPerBlock - 1) / wavesPerBlock;
    gemm_wmma_f32<<<grid, 256, 0, stream>>>(A, W, bias, C, M, K, N, lda);
}

extern "C" void kernel_launch(void* const* d_in, const int* in_sizes, int n_in,
                              void* d_out, int out_size, void* d_ws, size_t ws_size,
                              hipStream_t stream) {
    const float* enc_output  = (const float*)d_in[0];
    const int*   enc_len     = (const int*)  d_in[1];
    const int*   styles      = (const int*)  d_in[2];
    const int*   dec_in      = (const int*)  d_in[3];
    const int*   dec_out     = (const int*)  d_in[4];
    const float* emb_table   = (const float*)d_in[5];
    const float* style_table = (const float*)d_in[6];
    const float* proj_w      = (const float*)d_in[7];
    const float* proj_b      = (const float*)d_in[8];
    const float* w_ih        = (const float*)d_in[9];
    const float* b_ih        = (const float*)d_in[10];
    const float* w_hh        = (const float*)d_in[11];
    const float* b_hh        = (const float*)d_in[12];
    const float* out_w       = (const float*)d_in[13];
    const float* out_b       = (const float*)d_in[14];

    // workspace layout (floats)
    float* ws = (float*)d_ws;
    size_t off = 0;
    float* x  = ws + off; off += (size_t)kB * kT * kDIN;   // (B*T, DIN)
    float* a0 = ws + off; off += (size_t)kB * kKH0;        // (B, ENC+STY)
    float* x3 = ws + off; off += (size_t)kB * kT * kH3;    // (B*T, 3H)
    float* h  = ws + off; off += (size_t)kB * kH;          // (B, H)
    float* g  = ws + off; off += (size_t)kB * kH3;         // (B, 3H)
    float* hs = ws + off;                                  // (B, T, H) == (B*T, H)

    // output layout: [logits (B*T*V) | log_probs (B*T) | prediction (B*T)]
    float* logits = (float*)d_out;
    float* lp     = logits + (size_t)kB * kT * kV;
    float* pred   = lp + (size_t)kB * kT;

    // 1) gathers / concats
    {
        const size_t total = (size_t)kB * kT * kDIN + (size_t)kB * kKH0;
        const int grid = (int)((total + 255) / 256);
        prep_kernel<<<grid, 256, 0, stream>>>(enc_output, enc_len, styles, dec_in,
                                              emb_table, style_table, x, a0);
    }

    // 2) h0 = a0 @ proj_w^T + proj_b          (32 x 1152 x 1024)
    launch_gemm(a0, proj_w, proj_b, h, kB, kKH0, kH, kKH0, stream);

    // 3) x3 = x @ w_ih^T + b_ih               (2048 x 1664 x 3072)
    launch_gemm(x, w_ih, b_ih, x3, kB * kT, kDIN, kH3, kDIN, stream);

    // 4) GRU recurrence (sequential over T)
    for (int t = 0; t < kT; ++t) {
        // g = h @ w_hh^T + b_hh               (32 x 1024 x 3072)
        launch_gemm(h, w_hh, b_hh, g, kB, kH, kH3, kH, stream);
        gru_gate_kernel<<<(kB * kH + 255) / 256, 256, 0, stream>>>(x3, g, h, hs, t);
    }

    // 5) logits = hs @ out_w^T + out_b        (2048 x 1024 x 32000)
    launch_gemm(hs, out_w, out_b, logits, kB * kT, kH, kV, kH, stream);

    // 6) log-softmax stats, target gather, argmax
    logsoftmax_kernel<<<kB * kT, 256, 0, stream>>>(logits, dec_out, lp, pred);
}